// NCL_60730837565581
// MI455X (gfx1250) — compile-verified
//
#include <hip/hip_runtime.h>
#include <stdint.h>

#define USER_COUNT 100000
#define ITEM_COUNT 50000
#define N_NODES    (USER_COUNT + ITEM_COUNT)
#define EMBED      64
#define N_EDGES    2400000
#define BATCH      8192
#define N_LAYERS   3

typedef __attribute__((ext_vector_type(2))) float v2f;
typedef __attribute__((ext_vector_type(8))) float v8f;

// ---------------------------------------------------------------------------
// x = acc = concat(embed_user, embed_item), float4-vectorized
// ---------------------------------------------------------------------------
__global__ void k_init(const float4* __restrict__ eu, const float4* __restrict__ ei,
                       float4* __restrict__ x, float4* __restrict__ acc) {
    int i = blockIdx.x * blockDim.x + threadIdx.x;          // float4 index
    const int n4 = N_NODES * EMBED / 4;                      // 2,400,000
    if (i >= n4) return;
    const int u4 = USER_COUNT * EMBED / 4;                   // 1,600,000
    float4 v = (i < u4) ? eu[i] : ei[i - u4];
    x[i]   = v;
    acc[i] = v;
}

// ---------------------------------------------------------------------------
// vals = mask ? edge_vals / KEEP_PROB : 0
// ---------------------------------------------------------------------------
__global__ void k_vals(const float* __restrict__ ev, const uint8_t* __restrict__ mask,
                       float* __restrict__ vals) {
    int e = blockIdx.x * blockDim.x + threadIdx.x;
    if (e < N_EDGES)
        vals[e] = mask[e] ? ev[e] * (1.0f / 0.9f) : 0.0f;
}

__global__ void k_zero(float4* __restrict__ y) {
    int i = blockIdx.x * blockDim.x + threadIdx.x;
    if (i < N_NODES * EMBED / 4) y[i] = make_float4(0.f, 0.f, 0.f, 0.f);
}

__global__ void k_accum(float4* __restrict__ acc, const float4* __restrict__ y) {
    int i = blockIdx.x * blockDim.x + threadIdx.x;
    if (i >= N_NODES * EMBED / 4) return;
    float4 a = acc[i];
    float4 b = y[i];
    a.x += b.x; a.y += b.y; a.z += b.z; a.w += b.w;
    acc[i] = a;
}

// ---------------------------------------------------------------------------
// Edge scatter: y[row] += val * x[col]. One wave = 32 edges.
// Lanes load 32 edges' (row,col,val) coalesced, then broadcast each edge to
// SGPRs via readlane: per edge one global_load_b64 + two global_atomic_add_f32.
// All state stays L2-resident (x: 38.4MB, y: 38.4MB, edges: ~29MB << 192MB L2).
// ---------------------------------------------------------------------------
__global__ void k_scatter(const int* __restrict__ rows, const int* __restrict__ cols,
                          const float* __restrict__ vals, const float* __restrict__ x,
                          float* __restrict__ y) {
    const int lane   = threadIdx.x & 31;
    const int waveId = blockIdx.x * (blockDim.x >> 5) + (threadIdx.x >> 5);
    const int ebase  = waveId * 32;                          // N_EDGES % 32 == 0
    if (ebase >= N_EDGES) return;

    int   r = rows[ebase + lane];
    int   c = cols[ebase + lane];
    float v = vals[ebase + lane];
    unsigned vbits = __float_as_uint(v);

    #pragma unroll
    for (int i = 0; i < 32; ++i) {
        int   ri = __builtin_amdgcn_readlane(r, i);
        int   ci = __builtin_amdgcn_readlane(c, i);
        float vi = __uint_as_float((unsigned)__builtin_amdgcn_readlane((int)vbits, i));
        if (vi != 0.0f) {
            const float2 xv = *(const float2*)(x + ci * EMBED + lane * 2);
            float* dst = y + ri * EMBED + lane * 2;
            atomicAdd(dst,     vi * xv.x);
            atomicAdd(dst + 1, vi * xv.y);
        }
    }
}

// ---------------------------------------------------------------------------
// Final batched dot products via V_WMMA_F32_16X16X4_F32.
// One wave handles 16 pairs: D = EU(16x64) x EI^T(64x16), take diag, /16.
// A layout (16x4 f32): lanes 0-15 = M, v0 holds K0 (K2 on upper half-wave),
// v1 holds K1 (K3) -> per-lane float2 at row + k0 + 2*(lane>>4).
// B (4x16) mirrors with N on lanes.  D diag: lanes 0-7 vgpr m; lanes 24-31
// vgpr lane-24 hold m = lane-16.
// ---------------------------------------------------------------------------
__global__ void k_dot(const int* __restrict__ users, const int* __restrict__ items,
                      const float* __restrict__ acc, float* __restrict__ out) {
    const int lane = threadIdx.x & 31;
    const int tile = blockIdx.x * (blockDim.x >> 5) + (threadIdx.x >> 5); // 0..511
    const int ml   = lane & 15;
    const int half = lane >> 4;

    const int p  = tile * 16 + ml;                 // pair index, exact coverage
    const int u  = users[p];
    const int it = items[p];
    const float* ru = acc + u * EMBED;
    const float* ri = acc + (USER_COUNT + it) * EMBED;

    v8f d = {};
    #pragma unroll
    for (int k0 = 0; k0 < EMBED; k0 += 4) {
        v2f a = *(const v2f*)(ru + k0 + 2 * half);
        v2f b = *(const v2f*)(ri + k0 + 2 * half);
        d = __builtin_amdgcn_wmma_f32_16x16x4_f32(
                /*neg_a=*/false, a, /*neg_b=*/false, b,
                /*c_mod=*/(short)0, d, /*reuse_a=*/false, /*reuse_b=*/false);
    }

    // Diagonal extraction (after all WMMAs; EXEC was all-ones through the loop)
    int m = -1, rsel = 0;
    if (lane < 8)        { m = lane;        rsel = lane;      }
    else if (lane >= 24) { m = lane - 16;   rsel = lane - 24; }
    if (m >= 0) {
        float dv;
        switch (rsel) {
            case 0: dv = d[0]; break;  case 1: dv = d[1]; break;
            case 2: dv = d[2]; break;  case 3: dv = d[3]; break;
            case 4: dv = d[4]; break;  case 5: dv = d[5]; break;
            case 6: dv = d[6]; break;  default: dv = d[7]; break;
        }
        out[tile * 16 + m] = dv * (1.0f / 16.0f);   // fold (acc/4)·(acc/4)
    }
}

// ---------------------------------------------------------------------------
extern "C" void kernel_launch(void* const* d_in, const int* in_sizes, int n_in,
                              void* d_out, int out_size, void* d_ws, size_t ws_size,
                              hipStream_t stream) {
    const int*      users      = (const int*)d_in[0];
    const int*      items      = (const int*)d_in[1];
    const float*    embed_user = (const float*)d_in[2];
    const float*    embed_item = (const float*)d_in[3];
    const int*      edge_rows  = (const int*)d_in[4];
    const int*      edge_cols  = (const int*)d_in[5];
    const float*    edge_vals  = (const float*)d_in[6];
    const uint8_t*  edge_mask  = (const uint8_t*)d_in[7];

    float* ws  = (float*)d_ws;
    const int NX = N_NODES * EMBED;                  // 9.6M floats (38.4 MB)
    float* xA   = ws;
    float* xB   = ws + (size_t)NX;
    float* acc  = ws + (size_t)2 * NX;
    float* vals = ws + (size_t)3 * NX;               // +2.4M floats -> ~124.8 MB total

    const int n4      = NX / 4;                      // 2.4M float4
    const int blkN4   = (n4 + 255) / 256;
    const int blkEdge = (N_EDGES + 255) / 256;

    k_init<<<blkN4, 256, 0, stream>>>((const float4*)embed_user,
                                      (const float4*)embed_item,
                                      (float4*)xA, (float4*)acc);
    k_vals<<<blkEdge, 256, 0, stream>>>(edge_vals, edge_mask, vals);

    const int scatterBlocks = (N_EDGES / 32 + 7) / 8;   // 8 waves/block, 32 edges/wave
    float* xin = xA;
    float* xout = xB;
    for (int l = 0; l < N_LAYERS; ++l) {
        k_zero<<<blkN4, 256, 0, stream>>>((float4*)xout);
        k_scatter<<<scatterBlocks, 256, 0, stream>>>(edge_rows, edge_cols, vals, xin, xout);
        k_accum<<<blkN4, 256, 0, stream>>>((float4*)acc, (const float4*)xout);
        float* t = xin; xin = xout; xout = t;
    }

    // 8192 pairs / 16 per wave = 512 waves = 64 blocks x 8 waves
    k_dot<<<BATCH / 16 / 8, 256, 0, stream>>>(users, items, acc, (float*)d_out);
}